// PCQM_Net_41248865910791
// MI455X (gfx1250) — compile-verified
//
#include <hip/hip_runtime.h>
#include <hip/hip_bf16.h>

namespace {
constexpr int   N_    = 150000;
constexpr int   E_    = 600000;
constexpr int   G_    = 8192;
constexpr int   D_    = 128;
constexpr int   FIN_  = 16;
constexpr int   L_    = 4;
constexpr int   ELAB_ = 500000;
constexpr float BN_EPS_ = 1e-5f;
}

typedef float v2f __attribute__((ext_vector_type(2)));
typedef float v8f __attribute__((ext_vector_type(8)));

// ---------------------------------------------------------------------------
// Node encoder: x[N,16] @ enc_W[16,128] + enc_b  -> xf[N,128]
// ---------------------------------------------------------------------------
__global__ void k_encode(const float* __restrict__ x, const float* __restrict__ W,
                         const float* __restrict__ b, float* __restrict__ out) {
    __shared__ float xr[FIN_];
    const int n = blockIdx.x;
    const int t = threadIdx.x;
    if (t < FIN_) xr[t] = x[n * FIN_ + t];
    __syncthreads();
    float acc = b[t];
#pragma unroll
    for (int k = 0; k < FIN_; ++k) acc = fmaf(xr[k], W[k * D_ + t], acc);
    out[(size_t)n * D_ + t] = acc;
}

// ---------------------------------------------------------------------------
// WMMA fp32 GEMM: Y[M,128] = epilogue( X[M,128] @ W[128,128] )
// Block = 256 threads (8 waves); block tile = 64 rows x 128 cols.
// Each wave owns a 16-col strip and 4 M-subtiles, so every B fragment
// (4x16 slice of W, global/L0-resident) is reused by 4 WMMAs.
// A tile staged in LDS (64x132, pad kills bank conflicts, even kk keeps
// ds_load_b64 8B-aligned). K swept 128 in steps of 4 via
// V_WMMA_F32_16X16X4_F32 (full fp32 precision).
// Epilogue: optional input ReLU (on A load), bias, eval-BatchNorm, ReLU,
// accumulate-into-Y. Mrows must be a multiple of 16; tail subtiles are
// guarded uniformly (EXEC stays all-ones around WMMA).
// ---------------------------------------------------------------------------
template <bool RELU_IN, bool BN, bool RELU_OUT, bool ACCUM>
__global__ void __launch_bounds__(256) k_gemm128(
    const float* __restrict__ X, const float* __restrict__ W,
    const float* __restrict__ bias,
    const float* __restrict__ gam, const float* __restrict__ bet,
    const float* __restrict__ rmean, const float* __restrict__ rvar,
    float* __restrict__ Y, int Mrows) {
    constexpr int AS = 132;                 // LDS row stride (floats)
    __shared__ float sA[64 * AS];           // 33.8 KB
    const int rowBase = blockIdx.x << 6;
    const int t = threadIdx.x;

    // cooperative, coalesced float4 load of the 64x128 A tile (zero-pad tail)
#pragma unroll
    for (int j = 0; j < 8; ++j) {
        const int f4   = t + j * 256;       // float4 index 0..2047
        const int flat = f4 << 2;
        const int r    = flat >> 7;
        const int c    = flat & 127;
        float4 v = make_float4(0.f, 0.f, 0.f, 0.f);
        if (rowBase + r < Mrows)
            v = *(const float4*)(X + (size_t)(rowBase + r) * D_ + c);
        if (RELU_IN) {
            v.x = fmaxf(v.x, 0.f); v.y = fmaxf(v.y, 0.f);
            v.z = fmaxf(v.z, 0.f); v.w = fmaxf(v.w, 0.f);
        }
        *(float4*)(sA + r * AS + c) = v;    // AS*4 = 528 B (16B aligned rows)
    }
    __syncthreads();

    const int lane = t & 31;
    const int wv   = t >> 5;
    const int nIdx = lane & 15;             // A-row within subtile / C column lane
    const int half = lane >> 4;             // K-pair selector (ISA 16x4 f32 layout)
    const int cola = (wv << 4) + nIdx;      // global output column (0..127)
    const float* __restrict__ aBase = sA + nIdx * AS;

    v8f acc[4];
#pragma unroll
    for (int m = 0; m < 4; ++m) acc[m] = (v8f){0.f,0.f,0.f,0.f,0.f,0.f,0.f,0.f};

#pragma unroll
    for (int k = 0; k < D_; k += 4) {
        const int kk = k + (half << 1);
        v2f b;
        b.x = W[kk * D_ + cola];
        b.y = W[(kk + 1) * D_ + cola];
#pragma unroll
        for (int m = 0; m < 4; ++m) {
            const float* ap = aBase + m * (16 * AS) + kk;
            v2f a; a.x = ap[0]; a.y = ap[1];
            acc[m] = __builtin_amdgcn_wmma_f32_16x16x4_f32(
                /*neg_a=*/false, a, /*neg_b=*/false, b,
                /*c_mod=*/(short)0, acc[m], /*reuse_a=*/false, /*reuse_b=*/false);
        }
    }

    // per-column epilogue factors: y = acc*alpha + beta
    float alpha = 1.0f;
    float beta  = bias[cola];
    if (BN) {
        const float s = gam[cola] * rsqrtf(rvar[cola] + BN_EPS_);
        beta  = (beta - rmean[cola]) * s + bet[cola];
        alpha = s;
    }
#pragma unroll
    for (int m = 0; m < 4; ++m) {
        const int sub = rowBase + (m << 4);
        if (sub < Mrows) {                  // uniform guard (Mrows % 16 == 0)
#pragma unroll
            for (int r = 0; r < 8; ++r) {
                float y = fmaf(acc[m][r], alpha, beta);
                if (RELU_OUT) y = fmaxf(y, 0.0f);
                const size_t off = (size_t)(sub + r + (half << 3)) * D_ + cola;
                if (ACCUM) y += Y[off];
                Y[off] = y;
            }
        }
    }
}

// ---------------------------------------------------------------------------
// h = (1 + eps) * x_feat   (eps read from device gin_eps[i])
// ---------------------------------------------------------------------------
__global__ void k_scale(const float4* __restrict__ src, const float* __restrict__ epsP,
                        float4* __restrict__ dst, int n4) {
    const int i = blockIdx.x * blockDim.x + threadIdx.x;
    if (i >= n4) return;
    const float s = 1.0f + epsP[0];
    float4 v = src[i];
    v.x *= s; v.y *= s; v.z *= s; v.w *= s;
    dst[i] = v;
}

// ---------------------------------------------------------------------------
// h[dst] += x_feat[src]  — one wave per edge, 32 lanes x float4 = 512B row
// ---------------------------------------------------------------------------
__global__ void k_agg(const float* __restrict__ xf, const int* __restrict__ src,
                      const int* __restrict__ dst, float* __restrict__ h) {
    const unsigned tid = blockIdx.x * blockDim.x + threadIdx.x;  // E*32 threads exact
    const int e = tid >> 5;
    const int c = (tid & 31) << 2;
    const int s = src[e];
    const int t = dst[e];
    const float4 v = *(const float4*)(xf + (size_t)s * D_ + c);
    float* p = h + (size_t)t * D_ + c;
    unsafeAtomicAdd(p + 0, v.x);
    unsafeAtomicAdd(p + 1, v.y);
    unsafeAtomicAdd(p + 2, v.z);
    unsafeAtomicAdd(p + 3, v.w);
}

// ---------------------------------------------------------------------------
// Segment max pooling via int atomic trick; init -inf, finalize -inf -> 0
// ---------------------------------------------------------------------------
__global__ void k_fill_neginf(float* __restrict__ p, int n) {
    const int i = blockIdx.x * blockDim.x + threadIdx.x;
    if (i < n) p[i] = __int_as_float(0xFF800000);
}

__device__ __forceinline__ void atomicMaxF(float* addr, float v) {
    if (v >= 0.0f) atomicMax((int*)addr, __float_as_int(v));
    else           atomicMin((unsigned int*)addr, __float_as_uint(v));
}

__global__ void k_pool_max(const float* __restrict__ xf, const int* __restrict__ batch,
                           float* __restrict__ p) {
    const int i = blockIdx.x * blockDim.x + threadIdx.x;   // N*D threads exact
    const int n = i >> 7;
    const int d = i & 127;
    const float v = xf[i];
    const int g = batch[n];
    atomicMaxF(p + (size_t)g * D_ + d, v);
}

__global__ void k_pool_fin(float* __restrict__ p, int n) {
    const int i = blockIdx.x * blockDim.x + threadIdx.x;
    if (i < n && __float_as_uint(p[i]) == 0xFF800000u) p[i] = 0.0f;
}

// ---------------------------------------------------------------------------
// s[g] = out2[g,:] . post_W2 + post_b2
// ---------------------------------------------------------------------------
__global__ void k_post_dot(const float* __restrict__ out2, const float* __restrict__ w2,
                           const float* __restrict__ b2, float* __restrict__ s) {
    __shared__ float red[D_];
    const int g = blockIdx.x;
    const int t = threadIdx.x;
    red[t] = out2[(size_t)g * D_ + t] * w2[t];
    __syncthreads();
    for (int off = 64; off > 0; off >>= 1) {
        if (t < off) red[t] += red[t + off];
        __syncthreads();
    }
    if (t == 0) s[g] = red[0] + b2[0];
}

// ---------------------------------------------------------------------------
// pred[e] = s[eil[0,e]] * s[eil[1,e]];  second output = edge_label (as float)
// ---------------------------------------------------------------------------
__global__ void k_pred(const float* __restrict__ s, const int* __restrict__ eil,
                       const int* __restrict__ lab, float* __restrict__ o) {
    const int e = blockIdx.x * blockDim.x + threadIdx.x;
    if (e < ELAB_) {
        o[e] = s[eil[e]] * s[eil[ELAB_ + e]];
        o[ELAB_ + e] = (float)lab[e];
    }
}

// ---------------------------------------------------------------------------
extern "C" void kernel_launch(void* const* d_in, const int* in_sizes, int n_in,
                              void* d_out, int out_size, void* d_ws, size_t ws_size,
                              hipStream_t stream) {
    const float* x     = (const float*)d_in[0];
    const int*   ei    = (const int*)d_in[1];
    const int*   batch = (const int*)d_in[2];
    const int*   eil   = (const int*)d_in[3];
    const int*   elab  = (const int*)d_in[4];
    const float* encW  = (const float*)d_in[5];
    const float* encB  = (const float*)d_in[6];
    const float* initW = (const float*)d_in[7];
    const float* initB = (const float*)d_in[8];
    const float* geps  = (const float*)d_in[9];
    const float* W1  = (const float*)d_in[10];
    const float* b1  = (const float*)d_in[11];
    const float* g1  = (const float*)d_in[12];
    const float* be1 = (const float*)d_in[13];
    const float* rm1 = (const float*)d_in[14];
    const float* rv1 = (const float*)d_in[15];
    const float* W2  = (const float*)d_in[16];
    const float* b2  = (const float*)d_in[17];
    const float* g2  = (const float*)d_in[18];
    const float* be2 = (const float*)d_in[19];
    const float* rm2 = (const float*)d_in[20];
    const float* rv2 = (const float*)d_in[21];
    const float* linW = (const float*)d_in[22];
    const float* linB = (const float*)d_in[23];
    const float* pW1 = (const float*)d_in[24];
    const float* pB1 = (const float*)d_in[25];
    const float* pW2 = (const float*)d_in[26];
    const float* pB2 = (const float*)d_in[27];

    // workspace layout (floats)
    float* buf0   = (float*)d_ws;
    float* buf1   = buf0   + (size_t)N_ * D_;
    float* pooled = buf1   + (size_t)N_ * D_;
    float* outg   = pooled + (size_t)G_ * D_;
    float* out2   = outg   + (size_t)G_ * D_;
    float* sg     = out2   + (size_t)G_ * D_;

    const int ND = N_ * D_;                 // 19,200,000 (divisible by 256)
    const int GD = G_ * D_;                 // 1,048,576  (divisible by 256)
    const int gridN = (N_ + 63) / 64;       // 2344 (tail = 48 rows = 3 subtiles)
    const int gridG = G_ / 64;              // 128

    // x_feat = x @ enc_W + enc_b
    k_encode<<<N_, D_, 0, stream>>>(x, encW, encB, buf0);

    // out = pool_max(x_feat @ init_W + init_b)
    k_gemm128<false, false, false, false><<<gridN, 256, 0, stream>>>(
        buf0, initW, initB, nullptr, nullptr, nullptr, nullptr, buf1, N_);
    k_fill_neginf<<<GD / 256, 256, 0, stream>>>(outg, GD);
    k_pool_max<<<ND / 256, 256, 0, stream>>>(buf1, batch, outg);
    k_pool_fin<<<GD / 256, 256, 0, stream>>>(outg, GD);

    float* A = buf0;   // current x_feat
    float* B = buf1;   // scratch / next x_feat
    for (int i = 0; i < L_; ++i) {
        // h = (1+eps)*x_feat; then h[dst] += x_feat[src] over edges
        k_scale<<<(ND / 4) / 256, 256, 0, stream>>>(
            (const float4*)A, geps + i, (float4*)B, ND / 4);
        k_agg<<<(E_ * 32) / 256, 256, 0, stream>>>(A, ei, ei + E_, B);
        // h = relu(bn(h @ W1 + b1));  x_feat = relu(bn(h @ W2 + b2))
        k_gemm128<false, true, true, false><<<gridN, 256, 0, stream>>>(
            B, W1 + i * D_ * D_, b1 + i * D_,
            g1 + i * D_, be1 + i * D_, rm1 + i * D_, rv1 + i * D_, A, N_);
        k_gemm128<false, true, true, false><<<gridN, 256, 0, stream>>>(
            A, W2 + i * D_ * D_, b2 + i * D_,
            g2 + i * D_, be2 + i * D_, rm2 + i * D_, rv2 + i * D_, B, N_);
        // out += pool_max(x_feat) @ lin_W + lin_b
        k_fill_neginf<<<GD / 256, 256, 0, stream>>>(pooled, GD);
        k_pool_max<<<ND / 256, 256, 0, stream>>>(B, batch, pooled);
        k_pool_fin<<<GD / 256, 256, 0, stream>>>(pooled, GD);
        k_gemm128<false, false, false, true><<<gridG, 256, 0, stream>>>(
            pooled, linW + i * D_ * D_, linB + i * D_,
            nullptr, nullptr, nullptr, nullptr, outg, G_);
        float* t = A; A = B; B = t;         // new x_feat now in B -> becomes A
    }

    // post MLP: relu -> Linear -> relu -> Linear(D,1)
    k_gemm128<true, false, true, false><<<gridG, 256, 0, stream>>>(
        outg, pW1, pB1, nullptr, nullptr, nullptr, nullptr, out2, G_);
    k_post_dot<<<G_, D_, 0, stream>>>(out2, pW2, pB2, sg);

    // edge decode: pred[e] = s[i]*s[j]; append labels
    k_pred<<<(ELAB_ + 255) / 256, 256, 0, stream>>>(sg, eil, elab, (float*)d_out);
}